// NoTwoSpaces_14293651161380
// MI455X (gfx1250) — compile-verified
//
#include <hip/hip_runtime.h>
#include <stdint.h>

// Problem shape (reference): x is [4096, 8192] f32, scan along axis 1.
#define SEQ   8192
#define WAVES 8          // waves per workgroup (256 threads, wave32)
#define TILE  256        // elements staged per tile per row (1 KB)
#define NBUF  3          // triple-buffered async pipeline (2 tiles in flight)
#define NTILES (SEQ / TILE)

__global__ __launch_bounds__(WAVES * 32)
void no_two_scan_kernel(const float* __restrict__ x, float* __restrict__ out) {
    // Per-wave triple buffer: 8 waves * 3 * 256 floats = 24 KB LDS (of 320 KB).
    __shared__ float smem[WAVES][NBUF][TILE];

    const int lane = threadIdx.x & 31;
    const int wave = threadIdx.x >> 5;
    const int row  = blockIdx.x * WAVES + wave;

    const float* __restrict__ xrow = x   + (size_t)row * SEQ;
    float*       __restrict__ orow = out + (size_t)row * SEQ;

    // Issue one tile (256 floats = 1 KB) as two async b128 copies:
    // each lane moves 16 contiguous bytes global -> LDS (ASYNCcnt += 2).
    auto issue_tile = [&](int tile, int buf) {
        const float* g0 = xrow + tile * TILE + lane * 4;
        const float* g1 = g0 + 128;
        uint32_t l0 = (uint32_t)(uintptr_t)&smem[wave][buf][lane * 4];
        uint32_t l1 = l0 + 128 * 4;
        asm volatile("global_load_async_to_lds_b128 %0, %1, off"
                     :: "v"(l0), "v"(g0) : "memory");
        asm volatile("global_load_async_to_lds_b128 %0, %1, off"
                     :: "v"(l1), "v"(g1) : "memory");
    };

    // Prologue: two tiles in flight.
    issue_tile(0, 0);
    issue_tile(1, 1);

    bool sP = false;  // carry: was out[t-1] > thr (wave-uniform; init -inf => false)

    for (int t = 0; t < NTILES; ++t) {
        const int buf = t % NBUF;

        if (t + 2 < NTILES) {
            // Buffer (t+2)%NBUF was last read at iteration t-1; drain DS reads
            // before the async engine overwrites it.
            asm volatile("s_wait_dscnt 0x0" ::: "memory");
            issue_tile(t + 2, (t + 2) % NBUF);
            // 4 outstanding ops belong to tiles t+1, t+2 => tile t complete.
            asm volatile("s_wait_asynccnt 0x4" ::: "memory");
        } else if (t + 1 < NTILES) {
            // Only tile t+1 still in flight.
            asm volatile("s_wait_asynccnt 0x2" ::: "memory");
        } else {
            asm volatile("s_wait_asynccnt 0x0" ::: "memory");
        }

        const float* lp = &smem[wave][buf][0];
        const int tbase = t * TILE;

        #pragma unroll
        for (int j = 0; j < TILE / 32; ++j) {
            float v = lp[j * 32 + lane];                 // ds_load_b32, conflict-free
            bool pred = v > 0.5f;
            uint32_t m = __builtin_amdgcn_ballot_w32(pred);

            // Nearest g==0 position strictly below this lane (or virtual carry).
            uint32_t below = lane ? (0xFFFFFFFFu >> (32 - lane)) : 0u;
            uint32_t u = (~m) & below;
            // z = index of that zero; if none, virtual zero at -1 (carry false)
            // or -2 (carry true: prior element was an even-offset run member).
            int z = u ? (31 - __builtin_clz(u)) : (-1 - (int)sP);
            int offset = lane - z - 1;                   // offset within g-run
            float r = (pred && (offset & 1)) ? 0.0f : v; // zero odd-offset run members

            // Touch-once stream: non-temporal store, don't rinse L2.
            __builtin_nontemporal_store(r, &orow[tbase + j * 32 + lane]);

            // Uniform carry update: L = trailing all-ones run length of m.
            uint32_t notm = ~m;
            if (notm != 0) {
                int L = __builtin_clz(notm);             // 0..31 leading ones
                sP = (L & 1) != 0;                       // s[31] = L>0 && !((L-1)&1)
            }
            // notm == 0: run spans whole block, s alternates 32x -> sP unchanged.
        }
    }
}

extern "C" void kernel_launch(void* const* d_in, const int* in_sizes, int n_in,
                              void* d_out, int out_size, void* d_ws, size_t ws_size,
                              hipStream_t stream) {
    (void)n_in; (void)d_ws; (void)ws_size; (void)out_size;
    const float* x = (const float*)d_in[0];
    float* out = (float*)d_out;
    const int rows = in_sizes[0] / SEQ;                  // 4096
    dim3 grid(rows / WAVES), block(WAVES * 32);
    no_two_scan_kernel<<<grid, block, 0, stream>>>(x, out);
}